// Parallel_Linear_27462020891222
// MI455X (gfx1250) — compile-verified
//
#include <hip/hip_runtime.h>

typedef __attribute__((ext_vector_type(2))) float v2f;
typedef __attribute__((ext_vector_type(8))) float v8f;

#define B_DIM   64
#define CSEL    64
#define IN_DIM  256
#define OUT_DIM 256
#define KCHUNK  64
#define LDS_STRIDE (KCHUNK + 4)          // 68 floats; 68 % 64 == 4 -> conflict-free frag reads
#define TILE_F  (B_DIM * LDS_STRIDE)     // 4352 floats = 17408 B per tile (16B-aligned rows)

// One wave-wide async DMA of 16 bytes/lane: global memory -> LDS, no VGPR data,
// tracked by ASYNCcnt (CDNA5 GLOBAL_LOAD_ASYNC_TO_LDS_B128, GV addressing mode).
__device__ __forceinline__ void async_b128(unsigned lds_byte_off, const float* gaddr) {
    asm volatile("global_load_async_to_lds_b128 %0, %1, off"
                 :: "v"(lds_byte_off), "v"(gaddr) : "memory");
}

__global__ __launch_bounds__(256) void Parallel_Linear_wmma_f32(
    const float* __restrict__ x,        // (B, Csel, IN)
    const int*   __restrict__ channels, // (Csel,)
    const float* __restrict__ weight,   // (Ctot, OUT, IN)
    const float* __restrict__ bias,     // (Ctot, OUT)
    float*       __restrict__ out)      // (B, Csel, OUT)
{
    extern __shared__ __align__(16) float smem[];   // 4 tiles: A0,B0,A1,B1
    float* const Abuf[2] = { smem,              smem + 2 * TILE_F };
    float* const Bbuf[2] = { smem + TILE_F,     smem + 3 * TILE_F };

    const int c     = blockIdx.y;              // selected channel index 0..63
    const int nblk  = blockIdx.x;              // OUT quarter 0..3
    const int ch    = channels[c];             // gathered real channel
    const int tid   = threadIdx.x;
    const int wave  = tid >> 5;                // 0..7 (wave32)
    const int lane  = tid & 31;
    const int r     = lane & 15;
    const int halfk = (lane >> 4) << 1;        // 0 (lanes 0-15) or 2 (lanes 16-31)

    const int m_base = (wave >> 1) << 4;       // 0,16,32,48  (batch rows)
    const int n_off  = (wave & 1) << 5;        // 0 or 32     (within 64-wide block)
    const int n_base = nblk << 6;              // block's OUT base

    const float* xc  = x + ((size_t)c << 8);           // + c*IN
    const float* wch = weight + ((size_t)ch << 16);    // + ch*OUT*IN

    // Per-thread staging coordinates: each thread owns 4 rows x 16B for A and for B.
    const int c4   = (tid & 15) << 2;          // float offset within k-chunk (16B granules)
    const int row0 = tid >> 4;                 // 0..15

    auto stage = [&](int kb, int buf) {
        #pragma unroll
        for (int i = 0; i < 4; ++i) {          // x rows b = row0, row0+16, ...
            const int row = row0 + 16 * i;
            const float* ga = xc + (size_t)row * (CSEL * IN_DIM) + kb + c4;
            async_b128((unsigned)(uintptr_t)(Abuf[buf] + row * LDS_STRIDE + c4), ga);
        }
        #pragma unroll
        for (int i = 0; i < 4; ++i) {          // weight rows o = n_base + row
            const int row = row0 + 16 * i;
            const float* gb = wch + (((size_t)(n_base + row)) << 8) + kb + c4;
            async_b128((unsigned)(uintptr_t)(Bbuf[buf] + row * LDS_STRIDE + c4), gb);
        }
    };

    v8f acc0 = {};
    v8f acc1 = {};

    stage(0, 0);                               // prologue: chunk 0 in flight (8 async ops)

    int buf = 0;
    for (int kb = 0; kb < IN_DIM; kb += KCHUNK, buf ^= 1) {
        if (kb + KCHUNK < IN_DIM) {
            stage(kb + KCHUNK, buf ^ 1);       // prefetch next chunk (8 more async ops)
            // wait until only the next chunk's 8 ops remain -> current chunk landed
            asm volatile("s_wait_asynccnt 0x8" ::: "memory");
        } else {
            asm volatile("s_wait_asynccnt 0x0" ::: "memory");
        }
        __syncthreads();                       // cover other waves' async writes

        const float* LA = Abuf[buf];
        const float* LB = Bbuf[buf];
        #pragma unroll
        for (int ks = 0; ks < KCHUNK; ks += 4) {
            v2f a  = *(const v2f*)(LA + (m_base + r)     * LDS_STRIDE + ks + halfk);
            v2f b0 = *(const v2f*)(LB + (n_off + r)      * LDS_STRIDE + ks + halfk);
            v2f b1 = *(const v2f*)(LB + (n_off + 16 + r) * LDS_STRIDE + ks + halfk);
            acc0 = __builtin_amdgcn_wmma_f32_16x16x4_f32(
                       false, a, false, b0, (short)0, acc0, false, false);
            acc1 = __builtin_amdgcn_wmma_f32_16x16x4_f32(
                       false, a, false, b1, (short)0, acc1, false, false);
        }
        __syncthreads();                       // buffer reuse fence (refilled at kb+2)
    }

    // ---- bias + store (C/D layout: VGPR j -> M = j or j+8; N = lane%16) ----
    const float* bch   = bias + (size_t)ch * OUT_DIM;
    const float  bias0 = bch[n_base + n_off + r];
    const float  bias1 = bch[n_base + n_off + 16 + r];

    #pragma unroll
    for (int j = 0; j < 8; ++j) {
        const int m = m_base + ((lane < 16) ? j : (j + 8));
        const size_t o = ((size_t)m * CSEL + c) * OUT_DIM + n_base + n_off;
        out[o + r]      = acc0[j] + bias0;
        out[o + 16 + r] = acc1[j] + bias1;
    }
}

extern "C" void kernel_launch(void* const* d_in, const int* in_sizes, int n_in,
                              void* d_out, int out_size, void* d_ws, size_t ws_size,
                              hipStream_t stream) {
    (void)in_sizes; (void)n_in; (void)out_size; (void)d_ws; (void)ws_size;
    const float* x        = (const float*)d_in[0];
    const int*   channels = (const int*)  d_in[1];
    const float* weight   = (const float*)d_in[2];
    const float* bias     = (const float*)d_in[3];
    float*       out      = (float*)d_out;

    dim3 grid(OUT_DIM / 64, CSEL, 1);          // (4, 64)
    dim3 block(256, 1, 1);                     // 8 wave32 waves
    const size_t shmem = 4u * TILE_F * sizeof(float);   // 69632 B (double-buffered A+B)
    Parallel_Linear_wmma_f32<<<grid, block, shmem, stream>>>(x, channels, weight, bias, out);
}